// CorrelationRegularizer_86096914415914
// MI455X (gfx1250) — compile-verified
//
#include <hip/hip_runtime.h>
#include <stdint.h>

// Problem shape (from reference): w is [B, N] f32, N fixed at 2048, B derived.
#define N_COLS 2048
#define R_CHUNKS 64            // row chunks for pass-1 column stats
#define K1_THREADS 256
#define K3_THREADS 256
#define WAVES_PER_BLOCK (K3_THREADS / 32)
#define CHUNK_FLOATS 128       // 32 lanes * 4 floats per async b128 instruction
#define CHUNKS_PER_ROW (N_COLS / CHUNK_FLOATS)   // 16
#define NBUF 4                 // async pipeline depth (3 loads in flight)
#define AHEAD 3

// ---------------------------------------------------------------------------
// Pass 1: per-column partial sum and sum-of-squares over a chunk of rows.
// Grid: (N/256, R_CHUNKS), 256 threads. Coalesced: consecutive lanes read
// consecutive columns of a row. Streams the full 128 MiB once from HBM
// (populating the 192 MB L2 for pass 2).
// ---------------------------------------------------------------------------
__global__ __launch_bounds__(K1_THREADS)
void k1_colstats(const float* __restrict__ w, float* __restrict__ psum,
                 float* __restrict__ psq, int B) {
  const int col = blockIdx.x * K1_THREADS + threadIdx.x;
  const int chunk = blockIdx.y;
  const int rowsPer = B / R_CHUNKS;
  const float* p = w + (size_t)(chunk * rowsPer) * N_COLS + col;
  float s = 0.f, q = 0.f;
#pragma unroll 4
  for (int r = 0; r < rowsPer; ++r) {
    float v = p[(size_t)r * N_COLS];
    s += v;
    q += v * v;
  }
  psum[chunk * N_COLS + col] = s;
  psq[chunk * N_COLS + col] = q;
}

// ---------------------------------------------------------------------------
// Pass 1b: reduce chunk partials (fixed order -> deterministic), produce
// a_j = 1/std_j, c_j = -mean_j/std_j, and constants C = sum c, C2 = sum c^2.
// Single block of 256 threads.
// ---------------------------------------------------------------------------
__global__ __launch_bounds__(256)
void k2_finalize(const float* __restrict__ psum, const float* __restrict__ psq,
                 float* __restrict__ avec, float* __restrict__ cvec,
                 float* __restrict__ consts, int B) {
  __shared__ float rc[256], rc2[256];
  const float invB = 1.0f / (float)B;
  float accC = 0.f, accC2 = 0.f;
  for (int i = 0; i < N_COLS / 256; ++i) {
    const int col = i * 256 + threadIdx.x;
    float s = 0.f, q = 0.f;
    for (int r = 0; r < R_CHUNKS; ++r) {
      s += psum[r * N_COLS + col];
      q += psq[r * N_COLS + col];
    }
    const float mean = s * invB;
    const float var = q * invB - mean * mean;   // population variance
    const float aj = 1.0f / sqrtf(var);
    const float cj = -mean * aj;
    avec[col] = aj;
    cvec[col] = cj;
    accC += cj;
    accC2 += cj * cj;
  }
  rc[threadIdx.x] = accC;
  rc2[threadIdx.x] = accC2;
  __syncthreads();
  for (int s = 128; s > 0; s >>= 1) {
    if ((int)threadIdx.x < s) {
      rc[threadIdx.x] += rc[threadIdx.x + s];
      rc2[threadIdx.x] += rc2[threadIdx.x + s];
    }
    __syncthreads();
  }
  if (threadIdx.x == 0) {
    consts[0] = rc[0];
    consts[1] = rc2[0];
  }
}

// ---------------------------------------------------------------------------
// Pass 2: one wave32 per row. 4-deep CDNA5 async global->LDS pipeline
// (GLOBAL_LOAD_ASYNC_TO_LDS_B128 + ASYNCcnt): 3 x 512 B loads in flight per
// wave; async loads complete in order, so s_wait_asynccnt 3 guarantees the
// oldest chunk has landed. fs = a*w + c, so with u = a*w we accumulate
// t1 = sum u, t2 = sum u^2, t3 = sum u*c, then
//   s  = t1 + C,  s2 = t2 + 2*t3 + C2,  pm = (s^2 - s2) / (N(N-1)).
// ---------------------------------------------------------------------------
__global__ __launch_bounds__(K3_THREADS)
void k3_rowpass(const float* __restrict__ w, const float* __restrict__ avec,
                const float* __restrict__ cvec, const float* __restrict__ consts,
                float* __restrict__ blockPart, int B) {
  __shared__ __align__(16) float ac[2 * N_COLS];                          // 16 KiB
  __shared__ __align__(16) float tile[WAVES_PER_BLOCK][NBUF][CHUNK_FLOATS]; // 16 KiB
  __shared__ float wsum[WAVES_PER_BLOCK];

  // Stage a[] / c[] once per block (reused by every chunk of every row).
  for (int i = threadIdx.x; i < N_COLS; i += K3_THREADS) {
    ac[i] = avec[i];
    ac[N_COLS + i] = cvec[i];
  }
  __syncthreads();

  const float C = consts[0];
  const float C2 = consts[1];

  const int lane = threadIdx.x & 31;
  const int wv = threadIdx.x >> 5;
  const int row = blockIdx.x * WAVES_PER_BLOCK + wv;

  float acc = 0.0f;
  if (row < B) {
    const uint64_t gbase =
        (uint64_t)(uintptr_t)(w + (size_t)row * N_COLS) + (uint64_t)(lane * 16);
    // Low 32 bits of a flat shared pointer == LDS byte offset (aperture in
    // the high dword) -> usable as the async instruction's LDS address VGPR.
    const uint32_t lbase =
        (uint32_t)(uintptr_t)(&tile[wv][0][0]) + (uint32_t)(lane * 16);

    // Prologue: fill the pipeline with AHEAD chunks.
#pragma unroll
    for (int pre = 0; pre < AHEAD; ++pre) {
      uint32_t lp = lbase + (uint32_t)((pre % NBUF) * (CHUNK_FLOATS * 4));
      uint64_t gp = gbase + (uint64_t)(pre * (CHUNK_FLOATS * 4));
      asm volatile("global_load_async_to_lds_b128 %0, %1, off"
                   :: "v"(lp), "v"(gp) : "memory");
    }

    float t1 = 0.f, t2 = 0.f, t3 = 0.f;
#pragma unroll
    for (int ck = 0; ck < CHUNKS_PER_ROW; ++ck) {
      if (ck + AHEAD < CHUNKS_PER_ROW) {
        // Drain older LDS reads before overwriting a retired buffer, then
        // prefetch chunk ck+AHEAD; wait until <=AHEAD remain in flight
        // (in-order completion => chunk ck has landed).
        uint32_t ln =
            lbase + (uint32_t)(((ck + AHEAD) % NBUF) * (CHUNK_FLOATS * 4));
        uint64_t gn = gbase + (uint64_t)((ck + AHEAD) * (CHUNK_FLOATS * 4));
        asm volatile("s_wait_dscnt 0x0" ::: "memory");
        asm volatile("global_load_async_to_lds_b128 %0, %1, off"
                     :: "v"(ln), "v"(gn) : "memory");
        asm volatile("s_wait_asynccnt 0x3" ::: "memory");
      } else if (ck == CHUNKS_PER_ROW - 3) {
        asm volatile("s_wait_asynccnt 0x2" ::: "memory");
      } else if (ck == CHUNKS_PER_ROW - 2) {
        asm volatile("s_wait_asynccnt 0x1" ::: "memory");
      } else {
        asm volatile("s_wait_asynccnt 0x0" ::: "memory");
      }
      const float4 v = *(const float4*)(&tile[wv][ck % NBUF][lane * 4]);
      const int jb = ck * CHUNK_FLOATS + lane * 4;
      const float4 a4 = *(const float4*)(&ac[jb]);
      const float4 c4 = *(const float4*)(&ac[N_COLS + jb]);
      const float ux = a4.x * v.x, uy = a4.y * v.y;
      const float uz = a4.z * v.z, uw = a4.w * v.w;
      t1 += (ux + uy) + (uz + uw);
      t2 += (ux * ux + uy * uy) + (uz * uz + uw * uw);
      t3 += (ux * c4.x + uy * c4.y) + (uz * c4.z + uw * c4.w);
    }

    // wave32 butterfly reduction (deterministic).
    for (int off = 16; off > 0; off >>= 1) {
      t1 += __shfl_xor(t1, off, 32);
      t2 += __shfl_xor(t2, off, 32);
      t3 += __shfl_xor(t3, off, 32);
    }
    if (lane == 0) {
      const float s = t1 + C;
      const float s2 = t2 + 2.0f * t3 + C2;
      const float pm =
          (s * s - s2) * (1.0f / ((float)N_COLS * (float)(N_COLS - 1)));
      acc = fabsf(pm);
    }
  }
  if (lane == 0) wsum[wv] = acc;
  __syncthreads();
  if (threadIdx.x == 0) {
    float t = 0.f;
#pragma unroll
    for (int i = 0; i < WAVES_PER_BLOCK; ++i) t += wsum[i];
    blockPart[blockIdx.x] = t;
  }
}

// ---------------------------------------------------------------------------
// Final: deterministic reduce of per-block partials; out = L * mean_b |pm|.
// ---------------------------------------------------------------------------
__global__ __launch_bounds__(256)
void k4_final(const float* __restrict__ blockPart, int nPart,
              float* __restrict__ out, int B) {
  __shared__ float red[256];
  float t = 0.f;
  for (int i = threadIdx.x; i < nPart; i += 256) t += blockPart[i];
  red[threadIdx.x] = t;
  __syncthreads();
  for (int s = 128; s > 0; s >>= 1) {
    if ((int)threadIdx.x < s) red[threadIdx.x] += red[threadIdx.x + s];
    __syncthreads();
  }
  if (threadIdx.x == 0) out[0] = 1.0f /*L*/ * red[0] / (float)B;
}

extern "C" void kernel_launch(void* const* d_in, const int* in_sizes, int n_in,
                              void* d_out, int out_size, void* d_ws, size_t ws_size,
                              hipStream_t stream) {
  const float* w = (const float*)d_in[0];
  const int total = in_sizes[0];
  const int B = total / N_COLS;   // 16384 for the reference shape

  float* ws = (float*)d_ws;
  float* psum = ws;                              // R_CHUNKS * N
  float* psq = psum + R_CHUNKS * N_COLS;         // R_CHUNKS * N
  float* avec = psq + R_CHUNKS * N_COLS;         // N
  float* cvec = avec + N_COLS;                   // N
  float* consts = cvec + N_COLS;                 // 2
  float* blockPart = consts + 2;                 // B / WAVES_PER_BLOCK

  dim3 g1(N_COLS / K1_THREADS, R_CHUNKS);
  k1_colstats<<<g1, K1_THREADS, 0, stream>>>(w, psum, psq, B);
  k2_finalize<<<1, 256, 0, stream>>>(psum, psq, avec, cvec, consts, B);

  const int nb3 = (B + WAVES_PER_BLOCK - 1) / WAVES_PER_BLOCK;
  k3_rowpass<<<nb3, K3_THREADS, 0, stream>>>(w, avec, cvec, consts, blockPart, B);
  k4_final<<<1, 256, 0, stream>>>(blockPart, nb3, (float*)d_out, B);
}